// AUGRU_23261542875441
// MI455X (gfx1250) — compile-verified
//
#include <hip/hip_runtime.h>
#include <math.h>

// AUGRU: T=200, B=4096, D=64, H=64. Persistent-WG recurrence, f32 WMMA GEMMs,
// double-buffered GLOBAL_LOAD_ASYNC_TO_LDS x tiles, per-wave pre-swizzled weight
// fragments in LDS (all inner-loop DS loads use one base reg + immediate offsets).
typedef __attribute__((ext_vector_type(2))) float v2f;
typedef __attribute__((ext_vector_type(8))) float v8f;

#define T_STEPS 200
#define BATCH   4096
#define DIN     64
#define HID     64
#define G3      192
#define ROWS    16      // batch rows per workgroup
#define NTHREADS 64     // 2 waves (wave32)
#define STR     68      // padded LDS row stride (floats) for x/h tiles

#define KSTEPS  (DIN / 4)          // 16 WMMA K-steps
#define FRAGS   6                  // (3 gates) x (2 n-tiles) per wave
#define WPACK   (KSTEPS*FRAGS*64)  // floats per wave per matrix = 6144 (24 KB)

// LDS layout (float offsets)
#define OFF_X0    0
#define OFF_X1    (OFF_X0   + ROWS*STR)
#define OFF_H     (OFF_X1   + ROWS*STR)
#define OFF_WIHP  (OFF_H    + ROWS*STR)
#define OFF_WHHP  (OFF_WIHP + 2*WPACK)
#define OFF_BIH   (OFF_WHHP + 2*WPACK)
#define OFF_BHH   (OFF_BIH  + G3)
#define OFF_WT    (OFF_BHH  + G3)
#define OFF_LEN   (OFF_WT   + ROWS)
#define SMEM_FLOATS (OFF_LEN + ROWS)

__device__ __forceinline__ float sigmoid_(float x) {
    return 1.0f / (1.0f + __expf(-x));
}

// 4x GLOBAL_LOAD_ASYNC_TO_LDS_B128: 64B per thread, tracked by ASYNCcnt.
// vdst carries the LDS byte offset (flat->LDS mapping uses addr[31:0]).
__device__ __forceinline__ void async_tile_load(const float* g, uint32_t ldsoff) {
    #pragma unroll
    for (int j = 0; j < 4; ++j) {
        unsigned long long ga = (unsigned long long)(uintptr_t)(g + 4 * j);
        uint32_t la = ldsoff + 16u * j;
        asm volatile("global_load_async_to_lds_b128 %0, %1, off"
                     :: "v"(la), "v"(ga) : "memory");
    }
}
__device__ __forceinline__ void wait_async_le4() {
    asm volatile("s_wait_asynccnt 0x4" ::: "memory");
}
__device__ __forceinline__ void wait_async_0() {
    asm volatile("s_wait_asynccnt 0x0" ::: "memory");
}

__global__ void
__launch_bounds__(NTHREADS, 1)
__attribute__((amdgpu_waves_per_eu(1)))
augru_persistent_kernel(const float* __restrict__ x,
                        const float* __restrict__ attw,
                        const int*   __restrict__ lengths,
                        const float* __restrict__ W_ih,
                        const float* __restrict__ W_hh,
                        const float* __restrict__ b_ih,
                        const float* __restrict__ b_hh,
                        float*       __restrict__ out) {
    extern __shared__ float sm[];
    float* hS    = sm + OFF_H;
    float* wihP  = sm + OFF_WIHP;
    float* whhP  = sm + OFF_WHHP;
    float* bihS  = sm + OFF_BIH;
    float* bhhS  = sm + OFF_BHH;
    float* wtS   = sm + OFF_WT;
    float* lenS  = sm + OFF_LEN;

    const int tid  = threadIdx.x;
    const int wave = tid >> 5;
    const int lane = tid & 31;
    const int lh   = lane >> 4;
    const int ln   = lane & 15;
    const int kA   = 2 * lh;
    const int rowBase = blockIdx.x * ROWS;

    // ---- One-time staging (kept rolled: cold code, don't bloat I$) ----
    // Weights pre-swizzled into per-wave WMMA fragment order:
    //   P[w][ki][g*2+u][lane] = { W[ncol][4ki+2lh], W[ncol][4ki+2lh+1] },
    //   ncol = g*64 + w*32 + u*16 + (lane&15).
    {
        const int sl = tid >> 1;        // staging lane 0..31
        const int se = tid & 1;         // element within v2f
        const int slh = sl >> 4, sln = sl & 15;
        #pragma unroll 1
        for (int w = 0; w < 2; ++w) {
            #pragma unroll 1
            for (int ki = 0; ki < KSTEPS; ++ki) {
                #pragma unroll 1
                for (int gu = 0; gu < FRAGS; ++gu) {
                    const int g = gu >> 1, u = gu & 1;
                    const int ncol = g * HID + w * 32 + u * 16 + sln;
                    const int kk   = 4 * ki + 2 * slh + se;
                    const int dst  = w * WPACK + (ki * FRAGS + gu) * 64 + sl * 2 + se;
                    wihP[dst] = W_ih[ncol * DIN + kk];
                    whhP[dst] = W_hh[ncol * DIN + kk];
                }
            }
        }
    }
    #pragma unroll 1
    for (int c = tid; c < G3; c += NTHREADS) {
        bihS[c] = b_ih[c];
        bhhS[c] = b_hh[c];
    }
    #pragma unroll 1
    for (int c = tid; c < ROWS * STR; c += NTHREADS) hS[c] = 0.0f;
    if (tid < ROWS) lenS[tid] = (float)lengths[rowBase + tid];

    // ---- Async x-tile pipeline: thread -> (row = tid>>2, 16-col chunk) ----
    const int xr = tid >> 2;
    const int xc = (tid & 3) * 16;
    const float* gx = x + ((size_t)rowBase + xr) * DIN + xc;
    const uint32_t ldsoffA = (uint32_t)(uintptr_t)(sm + OFF_X0 + xr * STR + xc);
    const uint32_t ldsoffB = (uint32_t)(uintptr_t)(sm + OFF_X1 + xr * STR + xc);

    async_tile_load(gx, ldsoffA);          // prologue: t=0 tile into buffer 0
    gx += (size_t)BATCH * DIN;

    __syncthreads();

    // Single-base fragment pointers (all hot-loop DS offsets become immediates).
    const float* wiB  = wihP + wave * WPACK + lane * 2;
    const float* whB  = whhP + wave * WPACK + lane * 2;
    const float* hA   = hS + ln * STR + kA;
    const float* xA0  = sm + OFF_X0 + ln * STR + kA;
    const float* xA1  = sm + OFF_X1 + ln * STR + kA;

    // ---- Time loop (recurrence) ----
    for (int t = 0; t < T_STEPS; ++t) {
        if (t + 1 < T_STEPS) {
            async_tile_load(gx, (t & 1) ? ldsoffA : ldsoffB);
            gx += (size_t)BATCH * DIN;
        }
        if (tid < ROWS) wtS[tid] = attw[(size_t)t * BATCH + rowBase + tid];

        if (t + 1 < T_STEPS) wait_async_le4(); else wait_async_0();
        __syncthreads();

        const float* xA = (t & 1) ? xA1 : xA0;

        // ---- gi = x @ W_ih^T, gh = h @ W_hh^T via V_WMMA_F32_16X16X4_F32 ----
        v8f accI[FRAGS], accH[FRAGS];
        const v8f vz = {0.f, 0.f, 0.f, 0.f, 0.f, 0.f, 0.f, 0.f};
        #pragma unroll
        for (int f = 0; f < FRAGS; ++f) { accI[f] = vz; accH[f] = vz; }

        #pragma unroll
        for (int ki = 0; ki < KSTEPS; ++ki) {
            const v2f ax = *(const v2f*)(xA + 4 * ki);
            const v2f ah = *(const v2f*)(hA + 4 * ki);
            #pragma unroll
            for (int f = 0; f < FRAGS; ++f) {
                const int off = (ki * FRAGS + f) * 64;
                const v2f bi = *(const v2f*)(wiB + off);
                const v2f bh = *(const v2f*)(whB + off);
                accI[f] = __builtin_amdgcn_wmma_f32_16x16x4_f32(
                    false, ax, false, bi, (short)0, accI[f], false, false);
                accH[f] = __builtin_amdgcn_wmma_f32_16x16x4_f32(
                    false, ah, false, bh, (short)0, accH[f], false, false);
            }
        }

        // ---- Gates + state update (C/D layout: m = v + 8*lh, n = lane&15) ----
        // Fragment f = g*2+u  ->  r: accI[0+u], z: accI[2+u], n: accI[4+u].
        float hn[2][8];
        #pragma unroll
        for (int u = 0; u < 2; ++u) {
            const int n = wave * 32 + u * 16 + ln;
            const float bir = bihS[n],           bhr = bhhS[n];
            const float biz = bihS[HID + n],     bhz = bhhS[HID + n];
            const float bin = bihS[2 * HID + n], bhn = bhhS[2 * HID + n];
            #pragma unroll
            for (int v = 0; v < 8; ++v) {
                const int m = v + 8 * lh;
                const float r  = sigmoid_(accI[0 + u][v] + bir + accH[0 + u][v] + bhr);
                float       z  = sigmoid_(accI[2 + u][v] + biz + accH[2 + u][v] + bhz);
                const float nn = tanhf(accI[4 + u][v] + bin + r * (accH[4 + u][v] + bhn));
                z *= wtS[m];
                const float hold = hS[m * STR + n];
                const float hy   = (1.0f - z) * hold + z * nn;
                const bool alive = ((float)t < lenS[m]);
                hn[u][v] = alive ? hy : hold;
                out[((size_t)t * BATCH + rowBase + m) * HID + n] = alive ? hy : 0.0f;
            }
        }
        __syncthreads();
        #pragma unroll
        for (int u = 0; u < 2; ++u) {
            const int n = wave * 32 + u * 16 + ln;
            #pragma unroll
            for (int v = 0; v < 8; ++v)
                hS[(v + 8 * lh) * STR + n] = hn[u][v];
        }
        __syncthreads();
    }
}

extern "C" void kernel_launch(void* const* d_in, const int* in_sizes, int n_in,
                              void* d_out, int out_size, void* d_ws, size_t ws_size,
                              hipStream_t stream) {
    const float* x     = (const float*)d_in[0];
    const float* attw  = (const float*)d_in[1];
    const int*   lens  = (const int*)  d_in[2];
    const float* W_ih  = (const float*)d_in[3];
    const float* W_hh  = (const float*)d_in[4];
    const float* b_ih  = (const float*)d_in[5];
    const float* b_hh  = (const float*)d_in[6];
    float*       out   = (float*)d_out;

    const int smem_bytes = SMEM_FLOATS * (int)sizeof(float);   // ~110 KB (<320 KB/WGP)
    (void)hipFuncSetAttribute((const void*)augru_persistent_kernel,
                              hipFuncAttributeMaxDynamicSharedMemorySize, smem_bytes);

    dim3 grid(BATCH / ROWS);     // 256 persistent workgroups, 16 batch rows each
    dim3 block(NTHREADS);        // 2 waves (wave32)
    augru_persistent_kernel<<<grid, block, smem_bytes, stream>>>(
        x, attw, lens, W_ih, W_hh, b_ih, b_hh, out);
}